// NeuralODE_26577257628159
// MI455X (gfx1250) — compile-verified
//
#include <hip/hip_runtime.h>

#define NDIM 64
#define HDIM 128
#define BDIM 16
#define SDIM 8
#define RK 4

typedef __attribute__((ext_vector_type(2))) float v2f;
typedef __attribute__((ext_vector_type(8))) float v8f;

__device__ __forceinline__ v8f wmma4(v2f a, v2f b, v8f c) {
  // D = A(16x4,f32) * B(4x16,f32) + C(16x16,f32)
  return __builtin_amdgcn_wmma_f32_16x16x4_f32(false, a, false, b, (short)0, c,
                                               false, false);
}

__global__ void __launch_bounds__(256)
node_kernel(const float* __restrict__ Ys, const float* __restrict__ Ts,
            const float* __restrict__ W1, const float* __restrict__ b1,
            const float* __restrict__ vt, const float* __restrict__ W2,
            const float* __restrict__ b2, float* __restrict__ out) {
  // ---- LDS-resident working set (~179 KB, fits 320 KB/WGP) ----
  __shared__ float sW1[NDIM * HDIM];   // W1[j][k]
  __shared__ float sW1T[HDIM * NDIM];  // W1T[k][j]
  __shared__ float sW2[HDIM * NDIM];   // W2[k][i]
  __shared__ float sW2T[NDIM * HDIM];  // W2T[i][k]
  __shared__ float sb1[HDIM], svt[HDIM], sb2[NDIM];
  __shared__ float sH[BDIM * HDIM];    // tanh activations (16x128)
  __shared__ float sWv[BDIM * HDIM];   // w = sech2 * (a @ W2^T)
  __shared__ float sY[BDIM * NDIM];    // y state
  __shared__ float sA[BDIM * NDIM];    // adjoint state
  __shared__ float sYs[BDIM * NDIM];   // stage input y
  __shared__ float sAs[BDIM * NDIM];   // stage input a
  __shared__ float sF[BDIM * NDIM];    // dy stage output
  __shared__ float sDA[BDIM * NDIM];   // da stage output
  __shared__ float sYacc[BDIM * NDIM];
  __shared__ float sAacc[BDIM * NDIM];
  __shared__ float sGb1[HDIM], sGvt[HDIM], sGb2[NDIM];
  __shared__ float sTcur[BDIM], sDT[BDIM];

  const int tid = threadIdx.x;
  const int lane = tid & 31;
  const int wave = tid >> 5;
  const int l15 = lane & 15;
  const int kh = (lane >> 4) << 1;   // K sub-offset for A/B fragments
  const int mh = (lane >> 4) << 3;   // M sub-offset for C/D fragments

  // ---- load weights (+ transposed copies to avoid strided LDS reads) ----
  for (int e = tid; e < NDIM * HDIM; e += 256) {
    float w = W1[e];
    sW1[e] = w;
    sW1T[(e % HDIM) * NDIM + (e / HDIM)] = w;
    float w2 = W2[e];  // W2 is HDIM x NDIM, same element count
    sW2[e] = w2;
    sW2T[(e % NDIM) * HDIM + (e / NDIM)] = w2;
  }
  for (int e = tid; e < HDIM; e += 256) {
    sb1[e] = b1[e]; svt[e] = vt[e]; sGb1[e] = 0.f; sGvt[e] = 0.f;
  }
  for (int e = tid; e < NDIM; e += 256) { sb2[e] = b2[e]; sGb2[e] = 0.f; }

  v8f zero8 = {};
  v8f gw1acc[4], gw2acc[4];  // per-wave register grad tiles
#pragma unroll
  for (int i = 0; i < 4; ++i) { gw1acc[i] = zero8; gw2acc[i] = zero8; }
  __syncthreads();

  // ---- fragment loaders ----
  auto ldA = [&](const float* Mm, int ld, int kb) -> v2f {
    v2f a;
    a.x = Mm[l15 * ld + kb + kh];
    a.y = Mm[l15 * ld + kb + kh + 1];
    return a;
  };
  // transposed A with per-batch-row scale: A[M, K=b] = (dt_b*cw) * Mm[b][m0+l15]
  auto ldAT = [&](const float* Mm, int ld, int m0, int kb, float cw) -> v2f {
    int m = m0 + l15;
    v2f a;
    a.x = (sDT[kb + kh] * cw) * Mm[(kb + kh) * ld + m];
    a.y = (sDT[kb + kh + 1] * cw) * Mm[(kb + kh + 1) * ld + m];
    return a;
  };
  auto ldB = [&](const float* Mm, int ld, int kb, int n0) -> v2f {
    int n = n0 + l15;
    v2f b;
    b.x = Mm[(kb + kh) * ld + n];
    b.y = Mm[(kb + kh + 1) * ld + n];
    return b;
  };

  // ---- layer kernels ----
  // Z = sYs@W1 ; h = tanh(z + b1 + t*vt) -> sH     (all 8 waves)
  auto layer1 = [&](float toff) {
    int n0 = wave * 16;
    v8f acc = zero8;
#pragma unroll
    for (int kb = 0; kb < NDIM; kb += 4)
      acc = wmma4(ldA(sYs, NDIM, kb), ldB(sW1, HDIM, kb, n0), acc);
#pragma unroll
    for (int r = 0; r < 8; ++r) {
      int M = r + mh, Nc = n0 + l15;
      float ts = sTcur[M] + toff * sDT[M];
      sH[M * HDIM + Nc] = tanhf(acc[r] + sb1[Nc] + ts * svt[Nc]);
    }
  };
  // dy = sH@W2 + b2 -> sF                           (waves 0-3)
  auto layer2f = [&]() {
    int n0 = wave * 16;
    v8f acc = zero8;
#pragma unroll
    for (int kb = 0; kb < HDIM; kb += 4)
      acc = wmma4(ldA(sH, HDIM, kb), ldB(sW2, NDIM, kb, n0), acc);
#pragma unroll
    for (int r = 0; r < 8; ++r) {
      int M = r + mh, Nc = n0 + l15;
      sF[M * NDIM + Nc] = acc[r] + sb2[Nc];
    }
  };
  // u = sAs@W2T ; w = (1-h^2)*u -> sWv              (all 8 waves)
  auto layerU = [&]() {
    int n0 = wave * 16;
    v8f acc = zero8;
#pragma unroll
    for (int kb = 0; kb < NDIM; kb += 4)
      acc = wmma4(ldA(sAs, NDIM, kb), ldB(sW2T, HDIM, kb, n0), acc);
#pragma unroll
    for (int r = 0; r < 8; ++r) {
      int M = r + mh, Nc = n0 + l15;
      float h = sH[M * HDIM + Nc];
      sWv[M * HDIM + Nc] = (1.f - h * h) * acc[r];
    }
  };
  // da = -(sWv@W1T) -> sDA                          (waves 4-7)
  auto layerDA = [&]() {
    int n0 = (wave - 4) * 16;
    v8f acc = zero8;
#pragma unroll
    for (int kb = 0; kb < HDIM; kb += 4)
      acc = wmma4(ldA(sWv, HDIM, kb), ldB(sW1T, NDIM, kb, n0), acc);
#pragma unroll
    for (int r = 0; r < 8; ++r) {
      int M = r + mh, Nc = n0 + l15;
      sDA[M * NDIM + Nc] = -acc[r];
    }
  };

  // forward dynamics eval: sYs -> sF
  auto f_eval = [&](float toff) {
    layer1(toff);
    __syncthreads();
    if (wave < 4) layer2f();
    __syncthreads();
  };

  // augmented dynamics eval: (sYs,sAs) -> (sF,sDA) + grad accumulation
  // cw = -(stage_weight)/6 so c_b = sDT[b]*cw matches the adjoint sign.
  auto aug_eval = [&](float toff, float cw) {
    layer1(toff);
    __syncthreads();
    layerU();
    __syncthreads();
    if (wave < 4) layer2f(); else layerDA();
    // gW1 -= c * Ys^T @ Wv  : tiles (mt, nt=wave), K = batch(16)
#pragma unroll
    for (int mt = 0; mt < 4; ++mt) {
      v8f acc = gw1acc[mt];
#pragma unroll
      for (int kb = 0; kb < BDIM; kb += 4)
        acc = wmma4(ldAT(sYs, NDIM, mt * 16, kb, cw),
                    ldB(sWv, HDIM, kb, wave * 16), acc);
      gw1acc[mt] = acc;
    }
    // gW2 -= c * H^T @ As   : tiles (mt=wave, nt)
#pragma unroll
    for (int nt = 0; nt < 4; ++nt) {
      v8f acc = gw2acc[nt];
#pragma unroll
      for (int kb = 0; kb < BDIM; kb += 4)
        acc = wmma4(ldAT(sH, HDIM, wave * 16, kb, cw),
                    ldB(sAs, NDIM, kb, nt * 16), acc);
      gw2acc[nt] = acc;
    }
    // bias/vt/b2 gradients (column sums, wave-uniform thread ranges)
    if (tid < HDIM) {
      float s1 = 0.f, s2 = 0.f;
      for (int bb = 0; bb < BDIM; ++bb) {
        float cb = sDT[bb] * cw;
        float wv = sWv[bb * HDIM + tid];
        float ts = sTcur[bb] + toff * sDT[bb];
        s1 += cb * wv;
        s2 += cb * ts * wv;
      }
      sGb1[tid] += s1;
      sGvt[tid] += s2;
    } else if (tid < HDIM + NDIM) {
      int i = tid - HDIM;
      float s = 0.f;
      for (int bb = 0; bb < BDIM; ++bb)
        s += (sDT[bb] * cw) * sAs[bb * NDIM + i];
      sGb2[i] += s;
    }
    __syncthreads();
  };

  // ---- scan over time steps ----
  for (int s = 0; s < SDIM; ++s) {
    int sp = (s > 0) ? s - 1 : 0;
    // forward init: y0 = Ys[:, sp, :], t = t0, dt = (t1-t0)/RK
    for (int e = tid; e < BDIM * NDIM; e += 256)
      sY[e] = Ys[((e >> 6) * SDIM + sp) * NDIM + (e & 63)];
    if (tid < BDIM) {
      float t0v = Ts[tid * SDIM + sp];
      float t1v = Ts[tid * SDIM + s];
      sTcur[tid] = t0v;
      sDT[tid] = (t1v - t0v) * 0.25f;
    }
    __syncthreads();

    // ---- forward RK4 ----
    for (int sub = 0; sub < RK; ++sub) {
      for (int e = tid; e < BDIM * NDIM; e += 256) sYs[e] = sY[e];
      __syncthreads();
      f_eval(0.0f);  // k1
      for (int e = tid; e < BDIM * NDIM; e += 256) {
        float k = sF[e];
        sYacc[e] = k;
        sYs[e] = sY[e] + 0.5f * sDT[e >> 6] * k;
      }
      __syncthreads();
      f_eval(0.5f);  // k2
      for (int e = tid; e < BDIM * NDIM; e += 256) {
        float k = sF[e];
        sYacc[e] += 2.f * k;
        sYs[e] = sY[e] + 0.5f * sDT[e >> 6] * k;
      }
      __syncthreads();
      f_eval(0.5f);  // k3
      for (int e = tid; e < BDIM * NDIM; e += 256) {
        float k = sF[e];
        sYacc[e] += 2.f * k;
        sYs[e] = sY[e] + sDT[e >> 6] * k;
      }
      __syncthreads();
      f_eval(1.0f);  // k4
      for (int e = tid; e < BDIM * NDIM; e += 256)
        sY[e] += (sDT[e >> 6] * (1.f / 6.f)) * (sYacc[e] + sF[e]);
      if (tid < BDIM) sTcur[tid] += sDT[tid];
      __syncthreads();
    }

    // emit y1_hat, seed adjoint: a = 2*(y1_hat - y1), integrate t1 -> t0
    for (int e = tid; e < BDIM * NDIM; e += 256) {
      int r = e >> 6, j = e & 63;
      float yh = sY[e];
      out[(r * SDIM + s) * NDIM + j] = yh;
      sA[e] = 2.f * (yh - Ys[(r * SDIM + s) * NDIM + j]);
    }
    if (tid < BDIM) {
      float t0v = Ts[tid * SDIM + sp];
      float t1v = Ts[tid * SDIM + s];
      sTcur[tid] = t1v;
      sDT[tid] = (t0v - t1v) * 0.25f;
    }
    __syncthreads();

    // ---- adjoint RK4 (augmented state, grads accumulate in registers) ----
    for (int sub = 0; sub < RK; ++sub) {
      for (int e = tid; e < BDIM * NDIM; e += 256) {
        sYs[e] = sY[e];
        sAs[e] = sA[e];
      }
      __syncthreads();
      aug_eval(0.0f, -1.f / 6.f);  // k1
      for (int e = tid; e < BDIM * NDIM; e += 256) {
        float hdt = 0.5f * sDT[e >> 6];
        float ky = sF[e], ka = sDA[e];
        sYacc[e] = ky; sAacc[e] = ka;
        sYs[e] = sY[e] + hdt * ky;
        sAs[e] = sA[e] + hdt * ka;
      }
      __syncthreads();
      aug_eval(0.5f, -2.f / 6.f);  // k2
      for (int e = tid; e < BDIM * NDIM; e += 256) {
        float hdt = 0.5f * sDT[e >> 6];
        float ky = sF[e], ka = sDA[e];
        sYacc[e] += 2.f * ky; sAacc[e] += 2.f * ka;
        sYs[e] = sY[e] + hdt * ky;
        sAs[e] = sA[e] + hdt * ka;
      }
      __syncthreads();
      aug_eval(0.5f, -2.f / 6.f);  // k3
      for (int e = tid; e < BDIM * NDIM; e += 256) {
        float dt = sDT[e >> 6];
        float ky = sF[e], ka = sDA[e];
        sYacc[e] += 2.f * ky; sAacc[e] += 2.f * ka;
        sYs[e] = sY[e] + dt * ky;
        sAs[e] = sA[e] + dt * ka;
      }
      __syncthreads();
      aug_eval(1.0f, -1.f / 6.f);  // k4
      for (int e = tid; e < BDIM * NDIM; e += 256) {
        float c = sDT[e >> 6] * (1.f / 6.f);
        sY[e] += c * (sYacc[e] + sF[e]);
        sA[e] += c * (sAacc[e] + sDA[e]);
      }
      if (tid < BDIM) sTcur[tid] += sDT[tid];
      __syncthreads();
    }
  }

  // ---- write gradients ----
  float* oW1 = out + BDIM * SDIM * NDIM;
  float* ob1 = oW1 + NDIM * HDIM;
  float* ovt = ob1 + HDIM;
  float* oW2 = ovt + HDIM;
  float* ob2 = oW2 + HDIM * NDIM;
#pragma unroll
  for (int mt = 0; mt < 4; ++mt)
#pragma unroll
    for (int r = 0; r < 8; ++r)
      oW1[(mt * 16 + r + mh) * HDIM + (wave * 16 + l15)] = gw1acc[mt][r];
#pragma unroll
  for (int nt = 0; nt < 4; ++nt)
#pragma unroll
    for (int r = 0; r < 8; ++r)
      oW2[(wave * 16 + r + mh) * NDIM + (nt * 16 + l15)] = gw2acc[nt][r];
  if (tid < HDIM) { ob1[tid] = sGb1[tid]; ovt[tid] = sGvt[tid]; }
  if (tid >= HDIM && tid < HDIM + NDIM) ob2[tid - HDIM] = sGb2[tid - HDIM];
}

extern "C" void kernel_launch(void* const* d_in, const int* in_sizes, int n_in,
                              void* d_out, int out_size, void* d_ws,
                              size_t ws_size, hipStream_t stream) {
  (void)in_sizes; (void)n_in; (void)out_size; (void)d_ws; (void)ws_size;
  const float* Ys = (const float*)d_in[0];
  const float* Ts = (const float*)d_in[1];
  const float* W1 = (const float*)d_in[2];
  const float* b1 = (const float*)d_in[3];
  const float* vt = (const float*)d_in[4];
  const float* W2 = (const float*)d_in[5];
  const float* b2 = (const float*)d_in[6];
  node_kernel<<<1, 256, 0, stream>>>(Ys, Ts, W1, b1, vt, W2, b2, (float*)d_out);
}